// GCN_24283745091807
// MI455X (gfx1250) — compile-verified
//
#include <hip/hip_runtime.h>

typedef __attribute__((ext_vector_type(2))) float v2f;
typedef __attribute__((ext_vector_type(8))) float v8f;

#define KDIM   128   // D == H == 128 (K dimension of every GEMM)
#define MTILES 2     // 16-row tiles per wave (6250 row tiles / 2 = 3125 blocks, exact)

// ---------------- degree / normalization ----------------

__global__ __launch_bounds__(256) void k_deg_init(float* __restrict__ deg, int n) {
    int i = blockIdx.x * blockDim.x + threadIdx.x;
    if (i < n) deg[i] = 1.0f;                       // self-loop contributes 1 to every node
}

__global__ __launch_bounds__(256) void k_deg_edges(const int* __restrict__ dst,
                                                   float* __restrict__ deg, int e) {
    int i = blockIdx.x * blockDim.x + threadIdx.x;
    if (i < e) atomicAdd(&deg[dst[i]], 1.0f);
}

__global__ __launch_bounds__(256) void k_dinv(float* __restrict__ deg, int n) {
    int i = blockIdx.x * blockDim.x + threadIdx.x;
    if (i < n) {
        float d = deg[i];
        deg[i] = (d > 0.0f) ? rsqrtf(d) : 0.0f;     // deg >= 1 always, keep ref semantics
    }
}

// ---------------- dense GEMM via V_WMMA_F32_16X16X4_F32 ----------------
// out[N x NCOLS] = in[N x 128] @ w[128 x NCOLS], NCOLS compile-time.
// grid.x = ceil(N/16/MTILES), blockDim = (32, ceil(NCOLS/16)): one wave per
// (MTILES x 16 rows) x (16 cols) strip.
// A frag (16x4 f32, ISA 7.12.2): lane {lo=lane&15 -> M, hi=lane>>4}:
//   v0 = A[M][k0+2*hi], v1 = A[M][k0+2*hi+1]  -> one 8B load, immediate offset.
// B frag (4x16 f32): lane lo -> col, v0 = B[k0+2*hi][col], v1 = B[k0+2*hi+1][col].
//   GUARD case: column index is clamped (a WMMA B-column only feeds the same
//   D-column, so garbage in clamped lanes is dead) and only the store is guarded
//   -> inner loop is completely branch-free.
// C/D (16x16 f32): VGPR r -> row = r + 8*hi, col = lo.
template <int NCOLS, bool GUARD>
__global__ __launch_bounds__(256) void k_gemm_wmma(const float* __restrict__ in,
                                                   const float* __restrict__ w,
                                                   float* __restrict__ out,
                                                   int n) {
    const int lane = threadIdx.x;        // 0..31 (wave32)
    const int lo   = lane & 15;
    const int hi   = lane >> 4;
    const int row_base = blockIdx.x * (16 * MTILES);
    const int col      = threadIdx.y * 16 + lo;
    const int colc     = GUARD ? (col < NCOLS ? col : NCOLS - 1) : col;  // clamp, no mask

    // Per-tile A row base pointers (rows clamped so OOB tiles read safe memory;
    // their stores are suppressed below).
    const float* arow[MTILES];
#pragma unroll
    for (int t = 0; t < MTILES; ++t) {
        int r = row_base + t * 16 + lo;
        if (r > n - 1) r = n - 1;
        arow[t] = in + (size_t)r * KDIM + 2 * hi;    // +2*hi: this lane's K sub-pair
    }
    const float* brow = w + (size_t)(2 * hi) * NCOLS + colc;

    v8f c[MTILES];
#pragma unroll
    for (int t = 0; t < MTILES; ++t) c[t] = (v8f){};

#pragma unroll
    for (int k0 = 0; k0 < KDIM; k0 += 4) {
        v2f b;
        b.x = brow[(size_t)k0 * NCOLS];              // B[k0+2hi  ][col]
        b.y = brow[(size_t)(k0 + 1) * NCOLS];        // B[k0+2hi+1][col]
#pragma unroll
        for (int t = 0; t < MTILES; ++t) {
            v2f a = *(const v2f*)(arow[t] + k0);     // A[row][k0+2hi .. +1], 8B load
            c[t] = __builtin_amdgcn_wmma_f32_16x16x4_f32(
                       false, a, false, b, (short)0, c[t], false, false);
        }
    }

#pragma unroll
    for (int t = 0; t < MTILES; ++t) {
        if (row_base + t * 16 >= n) continue;        // wave-uniform branch
        if (GUARD && col >= NCOLS) continue;         // ragged columns: drop lanes
        float* orow = out + (size_t)(row_base + t * 16 + 8 * hi) * NCOLS + col;
#pragma unroll
        for (int r = 0; r < 8; ++r) {
            orow[(size_t)r * NCOLS] = c[t][r];       // row = base + t*16 + r + 8*hi
        }
    }
}

// ---------------- aggregation ----------------
// out[i][c] = hW[i][c] * dinv[i]^2  + bias[c]   (self-loop edge + bias, no atomics)
__global__ __launch_bounds__(256) void k_agg_init(const float* __restrict__ hw,
                                                  const float* __restrict__ dinv,
                                                  const float* __restrict__ bias,
                                                  float* __restrict__ out,
                                                  int n, int ncols) {
    const int nch4 = ncols >> 2;
    long long tid = (long long)blockIdx.x * blockDim.x + threadIdx.x;
    if (tid >= (long long)n * nch4) return;
    int i  = (int)(tid / nch4);
    int c4 = (int)(tid - (long long)i * nch4);
    float s = dinv[i];
    s = s * s;
    float4 v = ((const float4*)(hw + (size_t)i * ncols))[c4];
    float4 o;
    o.x = v.x * s + bias[c4 * 4 + 0];
    o.y = v.y * s + bias[c4 * 4 + 1];
    o.z = v.z * s + bias[c4 * 4 + 2];
    o.w = v.w * s + bias[c4 * 4 + 3];
    ((float4*)(out + (size_t)i * ncols))[c4] = o;
}

// out[dst[e]][:] += hW[src[e]][:] * dinv[src]*dinv[dst]   (fp32 atomics at L2)
__global__ __launch_bounds__(256) void k_scatter(const float* __restrict__ hw,
                                                 const float* __restrict__ dinv,
                                                 const int* __restrict__ src,
                                                 const int* __restrict__ dst,
                                                 float* __restrict__ out,
                                                 int e, int ncols) {
    const int nch4 = ncols >> 2;
    long long tid = (long long)blockIdx.x * blockDim.x + threadIdx.x;
    long long edge = tid / nch4;
    if (edge >= e) return;
    int c4 = (int)(tid - edge * nch4);
    int s = src[edge];
    int d = dst[edge];
    float wgt = dinv[s] * dinv[d];
    float4 v = ((const float4*)(hw + (size_t)s * ncols))[c4];
    float* o = out + (size_t)d * ncols + (size_t)c4 * 4;
    atomicAdd(o + 0, v.x * wgt);
    atomicAdd(o + 1, v.y * wgt);
    atomicAdd(o + 2, v.z * wgt);
    atomicAdd(o + 3, v.w * wgt);
}

__global__ __launch_bounds__(256) void k_relu(float* __restrict__ x, long long total) {
    long long i = (long long)blockIdx.x * blockDim.x + threadIdx.x;
    if (i < total) x[i] = fmaxf(x[i], 0.0f);
}

// ---------------- driver ----------------

extern "C" void kernel_launch(void* const* d_in, const int* in_sizes, int n_in,
                              void* d_out, int out_size, void* d_ws, size_t ws_size,
                              hipStream_t stream) {
    const float* x  = (const float*)d_in[0];
    const int*   ei = (const int*)d_in[1];
    const float* W0 = (const float*)d_in[2];
    const float* b0 = (const float*)d_in[3];
    const float* W1 = (const float*)d_in[4];
    const float* b1 = (const float*)d_in[5];
    const float* W2 = (const float*)d_in[6];
    const float* b2 = (const float*)d_in[7];

    const int n  = in_sizes[0] / KDIM;   // 100000
    const int e  = in_sizes[1] / 2;      // 1600000
    const int cc = 40;                   // n_classes (W2 is 128 x 40)
    const int* src = ei;                 // edge_index[0]
    const int* dst = ei + e;             // edge_index[1]

    float* ws   = (float*)d_ws;
    float* deg  = ws;                                  // n floats -> becomes dinv in place
    float* bufA = ws + (((size_t)n + 3) & ~(size_t)3); // GEMM outputs (N x 128)
    float* bufB = bufA + (size_t)n * KDIM;             // aggregated features (N x 128)
    float* outp = (float*)d_out;                       // N x cc logits

    const int rowTiles  = (n + 15) / 16;                       // 6250
    const int gemmBlks  = (rowTiles + MTILES - 1) / MTILES;    // 3125
    const dim3 gH(32, KDIM / 16);                              // 8 col tiles -> 128 cols
    const dim3 gC(32, (40 + 15) / 16);                         // 3 col tiles -> 40 cols

    // normalization
    k_deg_init <<<(n + 255) / 256, 256, 0, stream>>>(deg, n);
    k_deg_edges<<<(e + 255) / 256, 256, 0, stream>>>(dst, deg, e);
    k_dinv     <<<(n + 255) / 256, 256, 0, stream>>>(deg, n);

    const long long totH = (long long)n * (KDIM / 4);
    const long long edgH = (long long)e * (KDIM / 4);
    const long long totC = (long long)n * (cc / 4);
    const long long edgC = (long long)e * (cc / 4);

    // layer 0: bufA = x @ W0 ; bufB = A_hat * bufA + b0 ; relu
    k_gemm_wmma<128, false><<<dim3(gemmBlks), gH, 0, stream>>>(x, W0, bufA, n);
    k_agg_init <<<(unsigned)((totH + 255) / 256), 256, 0, stream>>>(bufA, deg, b0, bufB, n, KDIM);
    k_scatter  <<<(unsigned)((edgH + 255) / 256), 256, 0, stream>>>(bufA, deg, src, dst, bufB, e, KDIM);
    k_relu     <<<(unsigned)((totH * 4 + 255) / 256), 256, 0, stream>>>(bufB, (long long)n * KDIM);

    // layer 1: bufA = bufB @ W1 ; bufB = A_hat * bufA + b1 ; relu
    k_gemm_wmma<128, false><<<dim3(gemmBlks), gH, 0, stream>>>(bufB, W1, bufA, n);
    k_agg_init <<<(unsigned)((totH + 255) / 256), 256, 0, stream>>>(bufA, deg, b1, bufB, n, KDIM);
    k_scatter  <<<(unsigned)((edgH + 255) / 256), 256, 0, stream>>>(bufA, deg, src, dst, bufB, e, KDIM);
    k_relu     <<<(unsigned)((totH * 4 + 255) / 256), 256, 0, stream>>>(bufB, (long long)n * KDIM);

    // layer 2: bufA = bufB @ W2 (N x 40) ; out = A_hat * bufA + b2
    k_gemm_wmma<40, true><<<dim3(gemmBlks), gC, 0, stream>>>(bufB, W2, bufA, n);
    k_agg_init <<<(unsigned)((totC + 255) / 256), 256, 0, stream>>>(bufA, deg, b2, outp, n, cc);
    k_scatter  <<<(unsigned)((edgC + 255) / 256), 256, 0, stream>>>(bufA, deg, src, dst, outp, e, cc);
}